// LightGCN_87016037417239
// MI455X (gfx1250) — compile-verified
//
#include <hip/hip_runtime.h>
#include <stdint.h>

#define DIM 64
#define TPB 256        // 8 wave32 per block

#ifndef __has_builtin
#define __has_builtin(x) 0
#endif

#if __has_builtin(__builtin_amdgcn_global_load_async_to_lds_b32)
#define HAVE_ASYNC_LDS32 1
#else
#define HAVE_ASYNC_LDS32 0
#endif

#if __has_builtin(__builtin_amdgcn_s_wait_asynccnt)
#define HAVE_WAIT_ASYNC 1
#else
#define HAVE_WAIT_ASYNC 0
#endif

#define AS1 __attribute__((address_space(1)))
#define AS3 __attribute__((address_space(3)))

// One b32 id per lane: global -> LDS, async when available.
__device__ __forceinline__ void async_copy_id(const int* g, int* l) {
#if HAVE_ASYNC_LDS32
    __builtin_amdgcn_global_load_async_to_lds_b32((AS1 int*)g, (AS3 int*)l, 0, 0);
#else
    *l = *g;   // fallback: load + ds_store (compiler-tracked deps)
#endif
}

__device__ __forceinline__ void wait_async0() {
#if HAVE_ASYNC_LDS32
#if HAVE_WAIT_ASYNC
    __builtin_amdgcn_s_wait_asynccnt(0);
#else
    asm volatile("s_wait_asynccnt 0x0" ::: "memory");
#endif
    asm volatile("" ::: "memory");   // keep LDS reads after the wait
#endif
}

// ---------------------------------------------------------------------------
// Pull-based aggregation: one wave32 per destination node; lane l owns dims
// 2l,2l+1. For each 32-edge chunk of the node's in-edge list, the wave stages
// the source ids into its private LDS slice with one coalesced
// GLOBAL_LOAD_ASYNC_TO_LDS_B32 wave-op (ASYNCcnt), then accumulates the 256B
// source rows (one coalesced b64/lane gather per edge, next row prefetched)
// in registers. Fuses deg_inv scaling, h_next write, and the layer-mean
// accumulator update -> zero atomics, zero memsets in the hot path.
// ---------------------------------------------------------------------------
__global__ void gather_kernel(const unsigned* __restrict__ rowptr,
                              const int* __restrict__ csr_src,   // row*64, grouped by dst
                              const float* __restrict__ h,
                              float* __restrict__ hnext,
                              const float* __restrict__ deginv,
                              float* __restrict__ acc,
                              float post, int N) {
    __shared__ int ids[TPB];                  // 32 ids per wave
    const int lane = threadIdx.x & 31;
    const int wid  = threadIdx.x >> 5;
    const int node = blockIdx.x * (TPB / 32) + wid;   // wave-uniform
    if (node >= N) return;

    const unsigned beg = rowptr[node];
    const unsigned end = rowptr[node + 1];
    int* myids = &ids[wid * 32];

    float sx = 0.0f, sy = 0.0f;
    for (unsigned base = beg; base < end; base += 32u) {
        const int n = (int)min(32u, end - base);
        // csr_src is over-allocated by 32 entries: unpredicated stage is safe.
        async_copy_id(&csr_src[base + lane], &myids[lane]);
        wait_async0();
        int so = myids[0];
        for (int i = 0; i < n; ++i) {
            const int son = (i + 1 < n) ? myids[i + 1] : so;
            __builtin_prefetch(h + son + 2 * lane, 0, 0);    // next source row
            float2 v = ((const float2*)(h + so))[lane];      // 256B/wave gather
            sx += v.x;
            sy += v.y;
            so = son;
        }
    }

    const float s = deginv[node];
    float2 val = make_float2(sx * s, sy * s);
    ((float2*)(hnext + node * DIM))[lane] = val;

    float2* ap = (float2*)(acc + node * DIM) + lane;
    float2 a = *ap;
    a.x = (a.x + val.x) * post;
    a.y = (a.y + val.y) * post;
    *ap = a;
}

// ---------------------------------------------------------------------------
// CSR construction (done once, reused for all 3 layers).
// ---------------------------------------------------------------------------

// Degree histogram with native u32 atomics.
__global__ void deg_count_kernel(const long long* __restrict__ col,
                                 unsigned int* __restrict__ deg, int E) {
    int e = blockIdx.x * blockDim.x + threadIdx.x;
    if (e < E) atomicAdd(&deg[(int)col[e]], 1u);
}

// Per-256-chunk sums.
__global__ void partial_kernel(const unsigned* __restrict__ deg,
                               unsigned* __restrict__ partial, int N) {
    __shared__ unsigned s[256];
    int i = blockIdx.x * 256 + threadIdx.x;
    s[threadIdx.x] = (i < N) ? deg[i] : 0u;
    __syncthreads();
    for (int off = 128; off > 0; off >>= 1) {
        if (threadIdx.x < (unsigned)off) s[threadIdx.x] += s[threadIdx.x + off];
        __syncthreads();
    }
    if (threadIdx.x == 0) partial[blockIdx.x] = s[0];
}

// Single-block exclusive scan of the chunk sums (M ~ 600, one pass).
__global__ void scan_partials_kernel(const unsigned* __restrict__ partial,
                                     unsigned* __restrict__ blockOff, int M) {
    __shared__ unsigned s[1024];
    __shared__ unsigned carry;
    if (threadIdx.x == 0) carry = 0u;
    __syncthreads();
    for (int base = 0; base < M; base += 1024) {
        int i = base + (int)threadIdx.x;
        unsigned v = (i < M) ? partial[i] : 0u;
        s[threadIdx.x] = v;
        __syncthreads();
        for (int off = 1; off < 1024; off <<= 1) {
            unsigned t = (threadIdx.x >= (unsigned)off) ? s[threadIdx.x - off] : 0u;
            __syncthreads();
            s[threadIdx.x] += t;
            __syncthreads();
        }
        if (i < M) blockOff[i] = carry + s[threadIdx.x] - v;   // exclusive
        __syncthreads();
        if (threadIdx.x == 0) carry += s[1023];
        __syncthreads();
    }
}

// Per-chunk rescan: rowptr (exclusive), fill cursor, deg_inv as float.
__global__ void rowptr_kernel(const unsigned* __restrict__ deg,
                              const unsigned* __restrict__ blockOff,
                              unsigned* __restrict__ rowptr,
                              unsigned* __restrict__ cursor,
                              float* __restrict__ deginv, int N) {
    __shared__ unsigned s[256];
    int i = blockIdx.x * 256 + threadIdx.x;
    unsigned v = (i < N) ? deg[i] : 0u;
    s[threadIdx.x] = v;
    __syncthreads();
    for (int off = 1; off < 256; off <<= 1) {
        unsigned t = (threadIdx.x >= (unsigned)off) ? s[threadIdx.x - off] : 0u;
        __syncthreads();
        s[threadIdx.x] += t;
        __syncthreads();
    }
    if (i < N) {
        unsigned excl = blockOff[blockIdx.x] + s[threadIdx.x] - v;
        rowptr[i] = excl;
        cursor[i] = excl;
        if (i == N - 1) rowptr[N] = excl + v;
        deginv[i] = (v > 0u) ? (1.0f / (float)v) : 0.0f;
    }
}

// Scatter edges into CSR slots (2M one-time u32 atomics).
__global__ void fill_csr_kernel(const long long* __restrict__ row,
                                const long long* __restrict__ col,
                                unsigned* __restrict__ cursor,
                                int* __restrict__ csr_src, int E) {
    int e = blockIdx.x * blockDim.x + threadIdx.x;
    if (e < E) {
        int c = (int)col[e];
        unsigned pos = atomicAdd(&cursor[c], 1u);
        csr_src[pos] = (int)row[e] * DIM;
    }
}

// h = concat(user, item); acc(d_out) = same. float4 vectorized.
__global__ void init_x_kernel(const float4* __restrict__ uw,
                              const float4* __restrict__ iw,
                              float4* __restrict__ h,
                              float4* __restrict__ acc,
                              int nU4, int nT4) {
    int i = blockIdx.x * blockDim.x + threadIdx.x;
    if (i < nT4) {
        float4 v = (i < nU4) ? uw[i] : iw[i - nU4];
        h[i]   = v;
        acc[i] = v;
    }
}

// ---------------------------------------------------------------------------
// Host launch: graph-capture safe (only async ops on `stream`).
// ---------------------------------------------------------------------------
extern "C" void kernel_launch(void* const* d_in, const int* in_sizes, int n_in,
                              void* d_out, int out_size, void* d_ws, size_t ws_size,
                              hipStream_t stream) {
    const long long* edge = (const long long*)d_in[0];   // [2, E] int64
    const float*     uw   = (const float*)d_in[1];       // [nU, 64]
    const float*     iw   = (const float*)d_in[2];       // [nI, 64]

    const int E  = in_sizes[0] / 2;
    const int nU = in_sizes[1] / DIM;
    const int nI = in_sizes[2] / DIM;
    const int N  = nU + nI;
    const int M  = (N + 255) / 256;          // number of 256-chunks

    // Workspace: deg | partial | blockOff | rowptr | cursor | deginv | csr | hA | hB
    size_t off = 0;
    auto alloc = [&](size_t bytes) {
        void* p = (char*)d_ws + off;
        off += (bytes + 255) & ~(size_t)255;
        return p;
    };
    unsigned* deg      = (unsigned*)alloc((size_t)N * 4);
    unsigned* partial  = (unsigned*)alloc((size_t)M * 4);
    unsigned* blockOff = (unsigned*)alloc((size_t)M * 4);
    unsigned* rowptr   = (unsigned*)alloc((size_t)(N + 1) * 4);
    unsigned* cursor   = (unsigned*)alloc((size_t)N * 4);
    float*    deginv   = (float*)alloc((size_t)N * 4);
    int*      csr_src  = (int*)alloc((size_t)(E + 32) * 4);  // +32: unpredicated stage
    float*    hA       = (float*)alloc((size_t)N * DIM * 4);
    float*    hB       = (float*)alloc((size_t)N * DIM * 4);

    float* acc = (float*)d_out;                              // [N, 64] fp32

    // --- CSR build (once) ---
    (void)hipMemsetAsync(deg, 0, (size_t)N * 4, stream);
    deg_count_kernel<<<(E + TPB - 1) / TPB, TPB, 0, stream>>>(edge + E, deg, E);
    partial_kernel<<<M, 256, 0, stream>>>(deg, partial, N);
    scan_partials_kernel<<<1, 1024, 0, stream>>>(partial, blockOff, M);
    rowptr_kernel<<<M, 256, 0, stream>>>(deg, blockOff, rowptr, cursor, deginv, N);
    fill_csr_kernel<<<(E + TPB - 1) / TPB, TPB, 0, stream>>>(edge, edge + E, cursor, csr_src, E);

    // --- x into h and acc ---
    const int nU4 = nU * DIM / 4;
    const int nT4 = N * DIM / 4;
    init_x_kernel<<<(nT4 + TPB - 1) / TPB, TPB, 0, stream>>>(
        (const float4*)uw, (const float4*)iw, (float4*)hA, (float4*)acc, nU4, nT4);

    // --- 3 propagation layers, pull-based (no atomics, no memsets) ---
    float* cur = hA;
    float* nxt = hB;
    const int gatherBlocks = (N + (TPB / 32) - 1) / (TPB / 32);   // wave per node
    for (int l = 0; l < 3; ++l) {
        const float post = (l == 2) ? 0.25f : 1.0f;
        gather_kernel<<<gatherBlocks, TPB, 0, stream>>>(
            rowptr, csr_src, cur, nxt, deginv, acc, post, N);
        float* t = cur; cur = nxt; nxt = t;
    }
}